// DenoisingNet_1580547972055
// MI455X (gfx1250) — compile-verified
//
#include <hip/hip_runtime.h>
#include <stdint.h>

// out[batch, i] = ELU(W[0,i] + b[0]) broadcast over 65536 rows of 1024 floats.
// Pure store-bandwidth kernel: 256 MB written once, ~11 us at 23.3 TB/s.
//
// CDNA5 path: park the 4 KB row in LDS once, then stream it out with
// GLOBAL_STORE_ASYNC_FROM_LDS_B128 (ASYNCcnt-tracked async DMA path,
// LDS -> L2 -> HBM, no VGPR data traffic in the hot loop), with NT
// temporal hint since the 256 MB stream exceeds the 192 MB L2 and is
// written exactly once.

typedef float v4f __attribute__((ext_vector_type(4)));

__global__ __launch_bounds__(256) void denoise_broadcast_elu_async(
    const float* __restrict__ W,
    const float* __restrict__ b,
    float* __restrict__ out,
    int rows)
{
    constexpr int L = 1024;
    __shared__ v4f row[256];            // one full 1024-float row, 4 KB

    const int t   = threadIdx.x;        // 0..255
    const int col = t * 4;

    // ---- one-time tiny compute: row[col..col+3] = ELU(W + b) ----
    const float bias = b[0];
    v4f w = *(const v4f*)(W + col);
    v4f v;
#pragma unroll
    for (int k = 0; k < 4; ++k) {
        const float x = w[k] + bias;
        v[k] = (x > 0.0f) ? x : expm1f(x);   // jax.nn.elu, alpha = 1
    }
    row[t] = v;
    __syncthreads();   // s_wait_dscnt 0 + barrier: LDS visible to async engine

    // ---- streaming broadcast: async LDS->global B128 stores ----
    // Low 32 bits of a generic pointer to LDS are the LDS byte offset.
    const unsigned lds_off = (unsigned)(uintptr_t)&row[t];

    // GVS addressing: mem = SGPR64(out) + VGPR32(byte offset)
    unsigned goff = (unsigned)(((unsigned)blockIdx.x * (unsigned)L + (unsigned)col) * 4u);
    const unsigned stride = (unsigned)gridDim.x * (unsigned)L * 4u;  // bytes/step

    for (int r = blockIdx.x; r < rows; r += (int)gridDim.x) {
        asm volatile(
            "global_store_async_from_lds_b128 %0, %1, %2 th:TH_STORE_NT"
            :
            : "v"(goff), "v"(lds_off), "s"(out)
            : "memory");
        goff += stride;
    }

    // Ensure all async stores have landed before wave teardown.
    asm volatile("s_wait_asynccnt 0" ::: "memory");
}

extern "C" void kernel_launch(void* const* d_in, const int* in_sizes, int n_in,
                              void* d_out, int out_size, void* d_ws, size_t ws_size,
                              hipStream_t stream)
{
    // d_in[0] = input_list (unused by the reference math)
    // d_in[1] = W (1024 floats), d_in[2] = b (1 float)
    const float* W = (const float*)d_in[1];
    const float* b = (const float*)d_in[2];
    float* out     = (float*)d_out;

    constexpr int L = 1024;
    const int rows = out_size / L;   // 65536

    dim3 block(256);                 // 8 wave32s; one row per block
    dim3 grid(4096);                 // 16 rows per block

    hipLaunchKernelGGL(denoise_broadcast_elu_async, grid, block, 0, stream,
                       W, b, out, rows);
}